// test_ONNX_Match_86878598463820
// MI455X (gfx1250) — compile-verified
//
#include <hip/hip_runtime.h>
#include <hip/hip_bf16.h>

// NMS-match (mmcv nms_match semantics) for B=4, N=2048, C=4.
// Phase 1 (build_overlap_mask): per-wave 16x32 IoU tiles. The union base
//   (area_i + area_j) is computed with V_WMMA_F32_16X16X4_F32 as the outer
//   sum  [area_i, 1] x [[1],[area_j]]; intersection via VALU; iou>thr packed
//   into a [B][N][N/32] bitmask in d_ws (2 MB, L2 resident) via wave32 ballot
//   whose bit layout matches the WMMA C/D VGPR layout exactly.
// Phase 2 (greedy): one 256-thread workgroup per (b,c): bitonic argsort of
//   scores in LDS (stable tie-break on index, matching jnp.argsort), then the
//   sequential greedy scan reading 256B mask rows from L2.

typedef __attribute__((ext_vector_type(2))) float v2f;
typedef __attribute__((ext_vector_type(8))) float v8f;

#define NBOX 2048
#define NBATCH 4
#define NCLS 4
#define WORDS (NBOX / 32)   // 64 mask words per row
#define ROWT (NBOX / 16)    // 128 row tiles

__device__ __forceinline__ bool iou_gt(const float4 a, const float4 b,
                                       float sumArea, float thr) {
  // matches reference: wh = clip(rb - lt, 0); inter = w*h;
  // union = area_i + area_j - inter; iou = inter / max(union, 1e-6)
  float w = fminf(a.z, b.z) - fmaxf(a.x, b.x);
  float h = fminf(a.w, b.w) - fmaxf(a.y, b.y);
  w = fmaxf(w, 0.0f);
  h = fmaxf(h, 0.0f);
  float inter = w * h;
  float uni = sumArea - inter;
  float iou = inter / fmaxf(uni, 1e-6f);
  return iou > thr;
}

__global__ __launch_bounds__(256) void build_overlap_mask(
    const float4* __restrict__ boxes,   // [B][N] of (x1,y1,x2,y2)
    const float* __restrict__ thrP,     // [1]
    unsigned* __restrict__ mask)        // [B][N][WORDS]
{
  const int wid = blockIdx.x * (blockDim.x >> 5) + (threadIdx.x >> 5);
  const int b   = wid / (ROWT * WORDS);
  const int rem = wid % (ROWT * WORDS);
  const int ti  = rem / WORDS;
  const int tj  = rem % WORDS;
  const int i0  = ti * 16;
  const int j0  = tj * 32;

  const int lane = threadIdx.x & 31;
  const int l16  = lane & 15;
  const int half = lane >> 4;
  const float thr = thrP[0];

  const float4* bb = boxes + (size_t)b * NBOX;

  // ---- A fragment: 16x4, col0 = area_i, col1 = 1, cols 2..3 = 0 ----
  float4 abox = bb[i0 + l16];
  float areaA = (abox.z - abox.x) * (abox.w - abox.y);
  v2f afrag;
  afrag[0] = (lane < 16) ? areaA : 0.0f;   // K=0 (lanes 0-15) / K=2 (16-31)
  afrag[1] = (lane < 16) ? 1.0f  : 0.0f;   // K=1 / K=3

  // ---- column boxes: lane's column in each 16-wide half of the tile ----
  float4 cb0 = bb[j0 + l16];
  float4 cb1 = bb[j0 + 16 + l16];
  float areaB0 = (cb0.z - cb0.x) * (cb0.w - cb0.y);
  float areaB1 = (cb1.z - cb1.x) * (cb1.w - cb1.y);

  // ---- B fragments: 4x16, row0 = 1, row1 = area_j, rows 2..3 = 0 ----
  v2f bf0, bf1;
  bf0[0] = (lane < 16) ? 1.0f   : 0.0f;
  bf0[1] = (lane < 16) ? areaB0 : 0.0f;
  bf1[0] = (lane < 16) ? 1.0f   : 0.0f;
  bf1[1] = (lane < 16) ? areaB1 : 0.0f;

  v8f zero = {};
  // D[m][n] = area_i[m] + area_j[n]  (union base), one WMMA per 16-col half
  v8f d0 = __builtin_amdgcn_wmma_f32_16x16x4_f32(
      false, afrag, false, bf0, (short)0, zero, false, false);
  v8f d1 = __builtin_amdgcn_wmma_f32_16x16x4_f32(
      false, afrag, false, bf1, (short)0, zero, false, false);

  // ---- row boxes for this lane: rows i0 + 8*half + r, r = 0..7 ----
  float4 rbox[8];
#pragma unroll
  for (int r = 0; r < 8; ++r) rbox[r] = bb[i0 + 8 * half + r];

  unsigned mylo = 0u, myhi = 0u;
#pragma unroll
  for (int r = 0; r < 8; ++r) {
    // D layout: VGPR r, lanes 0-15 -> (M=r, N=lane), lanes 16-31 -> (M=r+8)
    bool p0 = iou_gt(rbox[r], cb0, d0[r], thr);
    bool p1 = iou_gt(rbox[r], cb1, d1[r], thr);
    unsigned b0 = __builtin_amdgcn_ballot_w32(p0);
    unsigned b1 = __builtin_amdgcn_ballot_w32(p1);
    // row i0+r   : cols j0..j0+15 from b0 lo, j0+16..j0+31 from b1 lo
    unsigned wlo = (b0 & 0xFFFFu) | (b1 << 16);
    // row i0+r+8 : from the hi halves
    unsigned whi = (b0 >> 16) | (b1 & 0xFFFF0000u);
    if (lane == r) { mylo = wlo; myhi = whi; }
  }

  if (lane < 8) {
    mask[((size_t)b * NBOX + i0 + lane)     * WORDS + tj] = mylo;
    mask[((size_t)b * NBOX + i0 + 8 + lane) * WORDS + tj] = myhi;
  }
}

__global__ __launch_bounds__(256) void nms_greedy(
    const float* __restrict__ scores,   // [B][C][N]
    const unsigned* __restrict__ mask,  // [B][N][WORDS]
    const float* __restrict__ sthrP,    // [1]
    int* __restrict__ out)              // [B][C][N]
{
  __shared__ float sKey[NBOX];
  __shared__ int   sIdx[NBOX];
  __shared__ int   sMatch[NBOX];
  __shared__ int   sSup[NBOX];

  const int tid = threadIdx.x;
  const int b = blockIdx.x / NCLS;
  const int c = blockIdx.x % NCLS;
  const float sthr = sthrP[0];
  const float* sc = scores + ((size_t)b * NCLS + c) * NBOX;

  for (int i = tid; i < NBOX; i += 256) {
    float s = sc[i];
    sKey[i] = s;
    sIdx[i] = i;
    sMatch[i] = -1;
    sSup[i] = (s <= sthr) ? 1 : 0;   // below score threshold: never matched
  }
  __syncthreads();

  // Bitonic sort, descending by score with index tie-break (stable-equivalent)
  for (int k = 2; k <= NBOX; k <<= 1) {
    for (int j = k >> 1; j > 0; j >>= 1) {
      for (int m = tid; m < NBOX; m += 256) {
        int p = m ^ j;
        if (p > m) {
          float ka = sKey[m], kb = sKey[p];
          int ia = sIdx[m], ib = sIdx[p];
          bool before = (ka > kb) || (ka == kb && ia < ib);
          bool up = ((m & k) == 0);
          if (up ? !before : before) {
            sKey[m] = kb; sKey[p] = ka;
            sIdx[m] = ib; sIdx[p] = ia;
          }
        }
      }
      __syncthreads();
    }
  }

  const unsigned* mbase = mask + (size_t)b * NBOX * WORDS;

  // Sequential greedy scan in score order
  for (int n = 0; n < NBOX; ++n) {
    int cand = sIdx[n];
    int active = (sSup[cand] == 0);
    __syncthreads();   // everyone reads sSup[cand] before anyone writes it
    if (active) {
      const unsigned* row = mbase + (size_t)cand * WORDS;
#pragma unroll
      for (int m2 = 0; m2 < NBOX / 256; ++m2) {
        int jb = tid + m2 * 256;   // each jb owned by exactly one thread
        if (!sSup[jb] && ((row[jb >> 5] >> (jb & 31)) & 1u)) {
          sMatch[jb] = cand;
          sSup[jb] = 1;
        }
      }
    }
    __syncthreads();   // make this round's suppression visible
  }

  int* o = out + ((size_t)b * NCLS + c) * NBOX;
  for (int i = tid; i < NBOX; i += 256) o[i] = sMatch[i];
}

extern "C" void kernel_launch(void* const* d_in, const int* in_sizes, int n_in,
                              void* d_out, int out_size, void* d_ws, size_t ws_size,
                              hipStream_t stream) {
  const float4* boxes   = (const float4*)d_in[0];   // [B,N,4] f32
  const float* scores   = (const float*)d_in[1];    // [B,C,N] f32
  const float* iou_thr  = (const float*)d_in[2];    // [1]
  const float* scr_thr  = (const float*)d_in[3];    // [1]
  int* out = (int*)d_out;                           // [B,C,N] int32
  unsigned* mask = (unsigned*)d_ws;                 // needs 2 MB scratch

  (void)in_sizes; (void)n_in; (void)out_size; (void)ws_size;

  // 32768 wave-tiles (B * 128 row-tiles * 64 col-words), 8 waves per block
  build_overlap_mask<<<(NBATCH * ROWT * WORDS) / 8, 256, 0, stream>>>(
      boxes, iou_thr, mask);
  // one workgroup per (batch, class)
  nms_greedy<<<NBATCH * NCLS, 256, 0, stream>>>(scores, mask, scr_thr, out);
}